// AttentionRNNModel_3126736191717
// MI455X (gfx1250) — compile-verified
//
#include <hip/hip_runtime.h>

typedef __attribute__((ext_vector_type(2))) float v2f;
typedef __attribute__((ext_vector_type(8))) float v8f;

#define B_ 128
#define T_ 512
#define D_ 64
#define H_ 1024
#define O_ 24
#define NBX 64           // H_/16 output-column tiles
#define LSTR 17          // LDS row stride (pad to avoid bank conflicts)

// One recurrence step, fused: h_t = tanh(x_t@W_i2h^T + h_{t-1}@W_h2h^T + b)
// plus accumulation of this step's contribution to the final FC.
// Grid: 64 blocks (ho-tile), 256 threads = 8 waves (b-tile per wave).
__global__ __launch_bounds__(256) void rnn_step_kernel(
    const float* __restrict__ x,     const float* __restrict__ W_i2h,
    const float* __restrict__ b_i2h, const float* __restrict__ W_h2h,
    const float* __restrict__ b_h2h, const float* __restrict__ W_fc,
    const float* __restrict__ h_prev, float* __restrict__ h_next,
    float* __restrict__ fcpart, int t, int first)
{
  __shared__ float tile[8 * 16 * LSTR];

  const int lane = threadIdx.x & 31;
  const int wave = threadIdx.x >> 5;   // b-tile 0..7
  const int bx   = blockIdx.x;         // ho-tile 0..63
  const int half = lane >> 4;          // K-half / M-half select
  const int lm   = lane & 15;

  const int brow = wave * 16 + lm;     // A-matrix row m (batch index)
  const int ncol = bx * 16 + lm;       // B-matrix col n (hidden-out index)

  v8f acc = {0.f, 0.f, 0.f, 0.f, 0.f, 0.f, 0.f, 0.f};

  // Recurrent GEMM: C += h_prev[b][hi] * W_h2h[ho][hi], K = 1024.
  // h0 == 0, so skip entirely on the first step.
  if (!first) {
    const float* arow = h_prev + (size_t)brow * H_;
    const float* bcol = W_h2h + (size_t)ncol * H_;
#pragma unroll 8
    for (int k = 0; k < H_; k += 4) {
      v2f a = *(const v2f*)(arow + k + half * 2);   // A[m][k..k+1]
      v2f b = *(const v2f*)(bcol + k + half * 2);   // B[k..k+1][n] = W[n][k..]
      acc = __builtin_amdgcn_wmma_f32_16x16x4_f32(
          false, a, false, b, (short)0, acc, false, false);
    }
  }

  // Input projection GEMM: C += x[b][t][d] * W_i2h[ho][d], K = 64.
  {
    const float* arow = x + (size_t)brow * (T_ * D_) + (size_t)t * D_;
    const float* bcol = W_i2h + (size_t)ncol * D_;
#pragma unroll
    for (int k = 0; k < D_; k += 4) {
      v2f a = *(const v2f*)(arow + k + half * 2);
      v2f b = *(const v2f*)(bcol + k + half * 2);
      acc = __builtin_amdgcn_wmma_f32_16x16x4_f32(
          false, a, false, b, (short)0, acc, false, false);
    }
  }

  // Epilogue: bias + tanh. D layout: acc[r] holds C[m = r + 8*half][n = lm].
  const float bias = b_i2h[ncol] + b_h2h[ncol];
  float hv[8];
#pragma unroll
  for (int r = 0; r < 8; ++r) {
    hv[r] = tanhf(acc[r] + bias);
    const int m = r + half * 8;
    h_next[(size_t)(wave * 16 + m) * H_ + ncol] = hv[r];
    tile[(wave * 16 + m) * LSTR + lm] = hv[r];
  }
  __syncthreads();

  // Fused FC partial: out[b][o] contribution of this 16x16 h tile.
  // Lane owns b_local = lm, o-range = [half*12, half*12+12).
  float hrow[16];
#pragma unroll
  for (int k = 0; k < 16; ++k)
    hrow[k] = tile[(wave * 16 + lm) * LSTR + k];

  float* fp = fcpart + ((size_t)bx * B_ + wave * 16 + lm) * O_;
  const float* wbase = W_fc + (size_t)t * H_ + (size_t)bx * 16;
#pragma unroll
  for (int oo = 0; oo < 12; ++oo) {
    const int o = half * 12 + oo;
    const float* wrow = wbase + (size_t)o * ((size_t)T_ * H_);
    float s = 0.f;
#pragma unroll
    for (int k = 0; k < 16; ++k) s += hrow[k] * wrow[k];
    fp[o] += s;  // exclusive ownership: one wave-lane per (bx,b,o); steps serialized
  }
}

__global__ void zero_kernel(float* __restrict__ p, int n) {
  int i = blockIdx.x * blockDim.x + threadIdx.x;
  if (i < n) p[i] = 0.f;
}

__global__ void fc_final_kernel(const float* __restrict__ fcpart,
                                const float* __restrict__ b_fc,
                                float* __restrict__ out) {
  int i = blockIdx.x * blockDim.x + threadIdx.x;
  if (i >= B_ * O_) return;
  const int o = i % O_;
  float s = b_fc[o];
#pragma unroll 8
  for (int bx = 0; bx < NBX; ++bx) s += fcpart[(size_t)bx * B_ * O_ + i];
  out[i] = s;
}

extern "C" void kernel_launch(void* const* d_in, const int* in_sizes, int n_in,
                              void* d_out, int out_size, void* d_ws, size_t ws_size,
                              hipStream_t stream) {
  const float* x     = (const float*)d_in[0];
  // d_in[1] = batchSize scalar (always 128 here)
  const float* W_i2h = (const float*)d_in[2];
  const float* b_i2h = (const float*)d_in[3];
  const float* W_h2h = (const float*)d_in[4];
  const float* b_h2h = (const float*)d_in[5];
  const float* W_fc  = (const float*)d_in[6];
  const float* b_fc  = (const float*)d_in[7];
  float* out = (float*)d_out;

  float* ws = (float*)d_ws;
  float* hbuf0  = ws;                    // [B, H]
  float* hbuf1  = ws + (size_t)B_ * H_;  // [B, H]
  float* fcpart = ws + (size_t)2 * B_ * H_;  // [NBX, B, O]

  // Zero the FC partial accumulators (fresh every call -> deterministic).
  {
    const int n = NBX * B_ * O_;
    zero_kernel<<<(n + 255) / 256, 256, 0, stream>>>(fcpart, n);
  }

  // 512 serial recurrence steps, ping-ponging the hidden state.
  float* hb[2] = {hbuf0, hbuf1};
  for (int t = 0; t < T_; ++t) {
    rnn_step_kernel<<<NBX, 256, 0, stream>>>(
        x, W_i2h, b_i2h, W_h2h, b_h2h, W_fc,
        hb[t & 1], hb[(t + 1) & 1], fcpart, t, (t == 0) ? 1 : 0);
  }

  // Reduce the 64 column-tile partials + bias into the final [B, O] output.
  fc_final_kernel<<<(B_ * O_ + 255) / 256, 256, 0, stream>>>(fcpart, b_fc, out);
}